// Attn_19464791785826
// MI455X (gfx1250) — compile-verified
//
#include <hip/hip_runtime.h>
#include <hip/hip_bf16.h>
#include <math.h>

// Sizes from the reference
#define BATCH  32
#define SEQ    4096
#define H      512
#define CHUNKS 32                  // L-splits per batch in the streaming kernel
#define LPW    (SEQ / CHUNKS / 8)  // tokens per wave = 16

typedef __attribute__((ext_vector_type(2))) float v2f;
typedef __attribute__((ext_vector_type(4))) float v4f;
typedef __attribute__((ext_vector_type(8))) float v8f;

// ---------------------------------------------------------------------------
// Kernel 1: V = hidden(32x512) @ W(512x512) using V_WMMA_F32_16X16X4_F32.
// One wave (32 threads) per 16x16 output tile. Grid = 2 (M tiles) * 32 (N tiles).
// A-tile layout (16x4 f32): lane&15 = m, (lane>>4)*2 = first of 2 K values.
// B-tile layout (4x16 f32): lane&15 = n, (lane>>4)*2 = first of 2 K values.
// C/D (16x16 f32, 8 VGPRs): row = r + 8*(lane>>4), col = lane&15.
// ---------------------------------------------------------------------------
__global__ void attn_vproj_wmma(const float* __restrict__ hidden,
                                const float* __restrict__ W,
                                float* __restrict__ V) {
    const int mt   = blockIdx.x & 1;    // 0..1  (M tile)
    const int nt   = blockIdx.x >> 1;   // 0..31 (N tile)
    const int lane = threadIdx.x;       // 0..31, wave32
    const int half = lane >> 4;         // 0 or 1
    const int idx  = lane & 15;

    const int m = mt * 16 + idx;        // row of hidden for A
    const int n = nt * 16 + idx;        // col of W for B

    v8f c = {};
    for (int k0 = 0; k0 < H; k0 += 4) {
        const int kk = k0 + half * 2;
        v2f a;
        a.x = hidden[m * H + kk];           // contiguous pair -> b64 load
        a.y = hidden[m * H + kk + 1];
        v2f b;
        b.x = W[kk * H + n];                // stride-H pair
        b.y = W[(kk + 1) * H + n];
        // (neg_a, A, neg_b, B, c_mod, C, reuse_a, reuse_b)
        c = __builtin_amdgcn_wmma_f32_16x16x4_f32(
                false, a, false, b, (short)0, c, false, false);
    }

#pragma unroll
    for (int r = 0; r < 8; ++r) {
        const int row = mt * 16 + r + 8 * half;
        V[row * H + nt * 16 + idx] = c[r];
    }
}

// ---------------------------------------------------------------------------
// Kernel 2: E[b,l] = enc[b,l,:] . V[b,:]   (bias dropped: cancels in softmax)
// Memory-bound: 256 MB of enc streamed once -> non-temporal loads (bigger
// than the 192 MB L2; zero reuse). Grid = 32 batches * 32 chunks, 256 threads
// (8 waves). Per token: 32 lanes each issue a b128 NT load (512 B contiguous
// per wave per load), 4 loads cover H=512, then a 5-step shfl_xor reduction.
// ---------------------------------------------------------------------------
__global__ void attn_energies(const float* __restrict__ enc,
                              const float* __restrict__ V,
                              float* __restrict__ E) {
    const int b     = blockIdx.x / CHUNKS;
    const int chunk = blockIdx.x % CHUNKS;

    __shared__ float vsh[H];
    for (int i = threadIdx.x; i < H; i += 256)
        vsh[i] = V[b * H + i];
    __syncthreads();

    const int wave = threadIdx.x >> 5;
    const int lane = threadIdx.x & 31;
    const float* encb = enc + (size_t)b * SEQ * H;
    const v4f*   vv   = (const v4f*)vsh;

    const int lbase = chunk * (SEQ / CHUNKS) + wave * LPW;
#pragma unroll 4
    for (int i = 0; i < LPW; ++i) {
        const int l = lbase + i;
        const v4f* row = (const v4f*)(encb + (size_t)l * H);
        float acc = 0.0f;
#pragma unroll
        for (int j = 0; j < 4; ++j) {
            const v4f e = __builtin_nontemporal_load(&row[lane + j * 32]);
            const v4f v = vv[lane + j * 32];
            acc += e.x * v.x + e.y * v.y + e.z * v.z + e.w * v.w;
        }
#pragma unroll
        for (int off = 16; off > 0; off >>= 1)
            acc += __shfl_xor(acc, off, 32);
        if (lane == 0)
            E[b * SEQ + l] = acc;
    }
}

// ---------------------------------------------------------------------------
// Kernel 3: out[b,0,:] = softmax(E[b,:]) over SEQ=4096. One block per batch,
// 256 threads * 16 values each. Shift-invariance already absorbed the bias.
// ---------------------------------------------------------------------------
__global__ void attn_softmax(const float* __restrict__ E,
                             float* __restrict__ out) {
    const int b = blockIdx.x;
    const int t = threadIdx.x;
    const int wave = t >> 5, lane = t & 31;

    __shared__ float redmax[8];
    __shared__ float redsum[8];

    const float* e = E + b * SEQ;
    float vals[16];
    float m = -INFINITY;
#pragma unroll
    for (int i = 0; i < 16; ++i) {
        vals[i] = e[t + i * 256];
        m = fmaxf(m, vals[i]);
    }
#pragma unroll
    for (int off = 16; off > 0; off >>= 1)
        m = fmaxf(m, __shfl_xor(m, off, 32));
    if (lane == 0) redmax[wave] = m;
    __syncthreads();
    float bm = redmax[0];
#pragma unroll
    for (int w = 1; w < 8; ++w) bm = fmaxf(bm, redmax[w]);

    float s = 0.0f;
#pragma unroll
    for (int i = 0; i < 16; ++i) {
        vals[i] = expf(vals[i] - bm);
        s += vals[i];
    }
#pragma unroll
    for (int off = 16; off > 0; off >>= 1)
        s += __shfl_xor(s, off, 32);
    if (lane == 0) redsum[wave] = s;
    __syncthreads();
    float bs = 0.0f;
#pragma unroll
    for (int w = 0; w < 8; ++w) bs += redsum[w];

    const float inv = 1.0f / bs;
#pragma unroll
    for (int i = 0; i < 16; ++i)
        out[b * SEQ + t + i * 256] = vals[i] * inv;
}

// ---------------------------------------------------------------------------
// Launch: inputs are (hidden, encoder_outputs, W, b). Bias is mathematically
// irrelevant under softmax, so it is never read. Workspace: V (64 KB) then
// E (512 KB).
// ---------------------------------------------------------------------------
extern "C" void kernel_launch(void* const* d_in, const int* in_sizes, int n_in,
                              void* d_out, int out_size, void* d_ws, size_t ws_size,
                              hipStream_t stream) {
    const float* hidden = (const float*)d_in[0];   // [1,32,512]
    const float* enc    = (const float*)d_in[1];   // [32,4096,512]
    const float* W      = (const float*)d_in[2];   // [512,512]
    // d_in[3] (bias) intentionally unused: softmax shift-invariance.

    float* V = (float*)d_ws;                       // [32,512]
    float* E = V + BATCH * H;                      // [32,4096]
    float* out = (float*)d_out;                    // [32,1,4096]

    attn_vproj_wmma<<<64, 32, 0, stream>>>(hidden, W, V);
    attn_energies<<<BATCH * CHUNKS, 256, 0, stream>>>(enc, V, E);
    attn_softmax<<<BATCH, 256, 0, stream>>>(E, out);
}